// SpinorBlock_52192442581441
// MI455X (gfx1250) — compile-verified
//
#include <hip/hip_runtime.h>

typedef float v2f __attribute__((ext_vector_type(2)));
typedef float v8f __attribute__((ext_vector_type(8)));
typedef int   v4i __attribute__((ext_vector_type(4)));

#define D_DIM   1024
#define NBLK    128
#define NTOK    16384               // B*S = 4*4096
#define OUT_HALF ((size_t)NTOK * D_DIM)
#define STRIDE  132                 // padded LDS token stride (floats): 128 data + 4 pad
#define TILES   16                  // 16-token tiles per workgroup

#if __has_builtin(__builtin_amdgcn_global_load_async_to_lds_b128)
#define USE_ASYNC 1
#else
#define USE_ASYNC 0
#endif

// Pointer types for the async builtin: pointee must be the v4i (b128) type.
// Constructed via integer round-trip (generic->int->ASn) to avoid illegal
// C++ address-space pointer casts.
typedef __attribute__((address_space(1))) v4i gas_v4i;
typedef __attribute__((address_space(3))) v4i lds_v4i;

__device__ __forceinline__ gas_v4i* to_gas(const void* p) {
  return (gas_v4i*)(unsigned long long)(uintptr_t)p;
}
__device__ __forceinline__ lds_v4i* to_lds(const void* p) {
  // flat LDS aperture: addr[31:0] is the LDS offset; AS3 pointers are 32-bit
  return (lds_v4i*)(unsigned int)(uintptr_t)p;
}

// ---------------------------------------------------------------------------
// Kernel 1: Cayley transform. One wave per 8x8 complex block; builds the
// 16x16 real-block system (I+A)U = (I-A) with A skew-symmetric, and runs
// Gauss-Jordan with one augmented-matrix column per lane (32 lanes = 32 cols).
// Writes M_n = U_rb (16x16, row-major) to workspace.
// ---------------------------------------------------------------------------
__global__ __launch_bounds__(32) void cayley_kernel(const float* __restrict__ Ar,
                                                    const float* __restrict__ Ai,
                                                    float* __restrict__ M) {
  const int n = blockIdx.x;
  const int j = threadIdx.x;            // augmented column: 0..15 LHS, 16..31 RHS
  const int c = j & 15;
  const float s = (j < 16) ? 1.0f : -1.0f;   // LHS = I + A, RHS = I - A
  const float* arn = Ar + n * 64;
  const float* ain = Ai + n * 64;
  const int c8 = c & 7, chi = c >> 3;

  float colv[16];
#pragma unroll
  for (int r = 0; r < 16; ++r) {
    const int r8 = r & 7, rhi = r >> 3;
    float ab;
    if (rhi == chi) {
      ab = 0.5f * (arn[r8 * 8 + c8] - arn[c8 * 8 + r8]);       // skew real part
    } else {
      const float ai = 0.5f * (ain[r8 * 8 + c8] + ain[c8 * 8 + r8]);
      ab = (rhi == 0) ? -ai : ai;                              // -Ai / +Ai
    }
    colv[r] = ((r == c) ? 1.0f : 0.0f) + s * ab;
  }

  // Gauss-Jordan, no pivoting (I + skew is well conditioned).
#pragma unroll
  for (int k = 0; k < 16; ++k) {
    const float piv = __shfl(colv[k], k, 32);
    const float pinv = 1.0f / piv;
    colv[k] *= pinv;
#pragma unroll
    for (int r = 0; r < 16; ++r) {
      if (r == k) continue;
      const float m = __shfl(colv[r], k, 32);
      colv[r] = __builtin_fmaf(-m, colv[k], colv[r]);
    }
  }

  if (j >= 16) {
    float* Mn = M + n * 256;
#pragma unroll
    for (int r = 0; r < 16; ++r) Mn[r * 16 + c] = colv[r];
  }
}

// ---------------------------------------------------------------------------
// Kernel 2: 128 GEMMs of 16x16 x 16x16384 via V_WMMA_F32_16X16X4_F32.
// 8 waves/WG = 8 consecutive blocks -> 256B-contiguous per-token traffic.
// X tiles double-buffered in LDS via GLOBAL_LOAD_ASYNC_TO_LDS_B128 (ASYNCcnt),
// overlapping tile t+1's HBM fetch with tile t's WMMA work.
// ---------------------------------------------------------------------------
__global__ __launch_bounds__(256) void apply_kernel(const float* __restrict__ xr,
                                                    const float* __restrict__ xi,
                                                    const float* __restrict__ M,
                                                    float* __restrict__ out) {
  __shared__ __align__(16) float Xb[2][16 * STRIDE];  // double-buffered X tile
  __shared__ __align__(16) float Ob[16 * STRIDE];     // output staging

  const int tid  = threadIdx.x;
  const int wave = tid >> 5;
  const int lane = tid & 31;
  const int half = lane >> 4;
  const int tk   = lane & 15;
  const int g    = blockIdx.x & 15;     // block group: blocks g*8 .. g*8+7
  const int chunk = blockIdx.x >> 4;
  const int n    = g * 8 + wave;

  // This thread's two (comp, token, quad) load slots, fixed across tiles.
  const int comp0 = tid >> 8, tt0 = (tid >> 4) & 15, q0 = tid & 15;
  const int idx1 = tid + 256;
  const int comp1 = idx1 >> 8, tt1 = (idx1 >> 4) & 15, q1 = idx1 & 15;

  // A-operand: M_n rows, lane = M (tk), VGPR r -> K = 4*kk + 2*half + r.
  v2f a0, a1, a2, a3;
  {
    const float* Mn = M + n * 256 + tk * 16 + 2 * half;
    a0 = *(const v2f*)(Mn + 0);
    a1 = *(const v2f*)(Mn + 4);
    a2 = *(const v2f*)(Mn + 8);
    a3 = *(const v2f*)(Mn + 12);
  }

#if USE_ASYNC
  // Issue one tile's X loads (8KB) straight into LDS: 2 async b128 per thread.
  auto issue_tile = [&](int t, int buf) {
    const int tok0 = (chunk * TILES + t) * 16;
    const float* s0 = (comp0 ? xi : xr) + (size_t)(tok0 + tt0) * D_DIM + g * 64 + q0 * 4;
    const float* s1 = (comp1 ? xi : xr) + (size_t)(tok0 + tt1) * D_DIM + g * 64 + q1 * 4;
    __builtin_amdgcn_global_load_async_to_lds_b128(
        to_gas(s0), to_lds(&Xb[buf][tt0 * STRIDE + comp0 * 64 + q0 * 4]), 0, 0);
    __builtin_amdgcn_global_load_async_to_lds_b128(
        to_gas(s1), to_lds(&Xb[buf][tt1 * STRIDE + comp1 * 64 + q1 * 4]), 0, 0);
  };

  issue_tile(0, 0);                       // prologue: tile 0 in flight

  for (int t = 0; t < TILES; ++t) {
    if (t + 1 < TILES) {
      issue_tile(t + 1, (t + 1) & 1);     // prefetch next tile (safe: that
                                          // buffer was last read before the
                                          // previous iteration's 2nd barrier)
      asm volatile("s_wait_asynccnt 0x2" ::: "memory");  // tile t landed
    } else {
      asm volatile("s_wait_asynccnt 0x0" ::: "memory");
    }
    __syncthreads();                      // all waves' tile-t data visible

    const float* Xt = Xb[t & 1];
    v8f acc = {0.f, 0.f, 0.f, 0.f, 0.f, 0.f, 0.f, 0.f};
#pragma unroll
    for (int kk = 0; kk < 4; ++kk) {
      const int K0   = 4 * kk + 2 * half;
      const int comp = K0 >> 3;
      const int rl   = K0 & 7;
      const v2f b = *(const v2f*)&Xt[tk * STRIDE + comp * 64 + wave * 8 + rl];
      const v2f a = (kk == 0) ? a0 : (kk == 1) ? a1 : (kk == 2) ? a2 : a3;
      acc = __builtin_amdgcn_wmma_f32_16x16x4_f32(
          false, a, false, b, (short)0, acc, false, false);
    }

    {
      float* o = &Ob[tk * STRIDE + half * 64 + wave * 8];
      *(float4*)(o + 0) = make_float4(acc[0], acc[1], acc[2], acc[3]);
      *(float4*)(o + 4) = make_float4(acc[4], acc[5], acc[6], acc[7]);
    }
    __syncthreads();                      // Ob visible to all waves

    const int tok0 = (chunk * TILES + t) * 16;
    {
      const float4 v0 = *(const float4*)&Ob[tt0 * STRIDE + comp0 * 64 + q0 * 4];
      *(float4*)(out + (size_t)comp0 * OUT_HALF + (size_t)(tok0 + tt0) * D_DIM +
                 g * 64 + q0 * 4) = v0;
      const float4 v1 = *(const float4*)&Ob[tt1 * STRIDE + comp1 * 64 + q1 * 4];
      *(float4*)(out + (size_t)comp1 * OUT_HALF + (size_t)(tok0 + tt1) * D_DIM +
                 g * 64 + q1 * 4) = v1;
    }
    // No 3rd barrier: Ob reads complete (data-dependent) before next
    // iteration's first barrier, after which Ob is rewritten.
  }
#else
  // Fallback: synchronous VGPR-staged loads.
  for (int t = 0; t < TILES; ++t) {
    const int tok0 = (chunk * TILES + t) * 16;
#pragma unroll
    for (int it = 0; it < 2; ++it) {
      const int idx = tid + it * 256;
      const int comp = idx >> 8, tt = (idx >> 4) & 15, q = idx & 15;
      const float* src = (comp ? xi : xr) + (size_t)(tok0 + tt) * D_DIM + g * 64 + q * 4;
      *(float4*)&Xb[0][tt * STRIDE + comp * 64 + q * 4] = *(const float4*)src;
    }
    __syncthreads();
    v8f acc = {0.f, 0.f, 0.f, 0.f, 0.f, 0.f, 0.f, 0.f};
#pragma unroll
    for (int kk = 0; kk < 4; ++kk) {
      const int K0 = 4 * kk + 2 * half;
      const int comp = K0 >> 3, rl = K0 & 7;
      const v2f b = *(const v2f*)&Xb[0][tk * STRIDE + comp * 64 + wave * 8 + rl];
      const v2f a = (kk == 0) ? a0 : (kk == 1) ? a1 : (kk == 2) ? a2 : a3;
      acc = __builtin_amdgcn_wmma_f32_16x16x4_f32(
          false, a, false, b, (short)0, acc, false, false);
    }
    {
      float* o = &Ob[tk * STRIDE + half * 64 + wave * 8];
      *(float4*)(o + 0) = make_float4(acc[0], acc[1], acc[2], acc[3]);
      *(float4*)(o + 4) = make_float4(acc[4], acc[5], acc[6], acc[7]);
    }
    __syncthreads();
#pragma unroll
    for (int it = 0; it < 2; ++it) {
      const int idx = tid + it * 256;
      const int comp = idx >> 8, tt = (idx >> 4) & 15, q = idx & 15;
      const float4 v = *(const float4*)&Ob[tt * STRIDE + comp * 64 + q * 4];
      *(float4*)(out + (size_t)comp * OUT_HALF + (size_t)(tok0 + tt) * D_DIM +
                 g * 64 + q * 4) = v;
    }
    __syncthreads();
  }
#endif
}

// ---------------------------------------------------------------------------
extern "C" void kernel_launch(void* const* d_in, const int* in_sizes, int n_in,
                              void* d_out, int out_size, void* d_ws, size_t ws_size,
                              hipStream_t stream) {
  const float* x_real = (const float*)d_in[0];
  const float* x_imag = (const float*)d_in[1];
  const float* A_real = (const float*)d_in[2];
  const float* A_imag = (const float*)d_in[3];
  float* out = (float*)d_out;
  float* Mws = (float*)d_ws;     // 128 blocks * 256 floats = 128 KB

  cayley_kernel<<<dim3(NBLK), dim3(32), 0, stream>>>(A_real, A_imag, Mws);
  // 16 block-groups x 64 token-chunks; each workgroup does 16 tiles of 16 tokens
  apply_kernel<<<dim3(16 * 64), dim3(256), 0, stream>>>(x_real, x_imag, Mws, out);
}